// KVGather_13529146982481
// MI455X (gfx1250) — compile-verified
//
#include <hip/hip_runtime.h>

// Problem constants from the reference:
//   N=16, P2=64, TOPK=8, W2=64, C_KV=128
// out[n,i,k,w,c] = r_weight[n,i,k] * kv[n, r_idx[n,i,k], w, c]
// Pure gather+scale: store-bandwidth bound (~268 MB out @ 23.3 TB/s ~= 13 us).

typedef __attribute__((ext_vector_type(4))) float v4f;

constexpr int P2    = 64;
constexpr int TOPK  = 8;
constexpr int W2    = 64;
constexpr int C_KV  = 128;
constexpr int TILE  = W2 * C_KV;            // 8192 floats = 32 KB per gathered slab
constexpr int THREADS = 256;                // 8 wave32s per block
constexpr int V4_PER_TILE = TILE / 4;       // 2048 float4s
constexpr int ITERS = V4_PER_TILE / THREADS;// 8 b128 load/store pairs per thread

__global__ __launch_bounds__(THREADS) void kvgather_scale_kernel(
    const int*   __restrict__ r_idx,    // (N, P2, TOPK) flat, values in [0, P2)
    const float* __restrict__ r_weight, // (N, P2, TOPK) flat
    const float* __restrict__ kv,       // (N, P2, W2, C_KV) flat
    float*       __restrict__ out)      // (N, P2, TOPK, W2, C_KV) flat
{
    // One block per (n, i, k) output tile.
    const int tile = blockIdx.x;                  // = (n*P2 + i)*TOPK + k
    const int n    = tile / (P2 * TOPK);

    // Block-uniform scalars -> expect s_load_b32 (KMcnt path).
    const int   idx = r_idx[tile];
    const float w   = r_weight[tile];

    // Source slab: contiguous 32 KB, read temporal (RT) so the ~8x per-slab
    // reuse within a batch n is served from the 192 MB L2 (kv is 33.5 MB total).
    const v4f* __restrict__ src =
        (const v4f*)(kv + (size_t)(n * P2 + idx) * (size_t)TILE);
    // Dest slab: contiguous 32 KB of a 268 MB streaming output -> NT stores so
    // we don't evict the hot kv working set from L2.
    v4f* __restrict__ dst = (v4f*)(out + (size_t)tile * (size_t)TILE);

    const int t = threadIdx.x;
#pragma unroll
    for (int j = 0; j < ITERS; ++j) {
        v4f v = src[j * THREADS + t];   // global_load_b128, coalesced
        v *= w;                         // packed f32 scale
        __builtin_nontemporal_store(v, &dst[j * THREADS + t]); // global_store_b128 th:NT
    }
}

extern "C" void kernel_launch(void* const* d_in, const int* in_sizes, int n_in,
                              void* d_out, int out_size, void* d_ws, size_t ws_size,
                              hipStream_t stream) {
    const int*   r_idx    = (const int*)  d_in[0]; // (N,P2,TOPK) int
    const float* r_weight = (const float*)d_in[1]; // (N,P2,TOPK) f32
    const float* kv       = (const float*)d_in[2]; // (N,P2,W2,C_KV) f32
    float*       out      = (float*)      d_out;   // (N,P2,TOPK,W2,C_KV) f32

    const int n_batch = in_sizes[0] / (P2 * TOPK); // N (=16)
    const int n_tiles = n_batch * P2 * TOPK;       // 8192 blocks

    kvgather_scale_kernel<<<dim3(n_tiles), dim3(THREADS), 0, stream>>>(
        r_idx, r_weight, kv, out);
}